// StateCell_79474074845629
// MI455X (gfx1250) — compile-verified
//
#include <hip/hip_runtime.h>
#include <math.h>

#define D   2048
#define D2  4096
#define NS  8
#define B   16384

typedef __attribute__((ext_vector_type(2))) float v2f;
typedef __attribute__((ext_vector_type(8))) float v8f;

// workspace layout (float offsets)
#define WS_M    0                 // [D*NS]  M[k][s] = sum_d keys[s][d]*Wr[d][k]
#define WS_S    (D*NS)            // [NS*D]  S = state + fast
#define WS_W0   (2*D*NS)          // [NS*D]  attn.T @ (h+neighbor_msg)   (atomic accum)
#define WS_GRAM (3*D*NS)          // [64]    attn.T @ attn               (atomic accum)
#define WS_COL  (WS_GRAM+64)      // [8]     colsum(attn)                (atomic accum)
#define WS_GATE (WS_COL+8)        // [1]     sum sigmoid(gate)           (atomic accum)
#define WS_U    (WS_GATE+1)       // [8]     u[s] = S[s] . Wc[0][D:2D]
#define WS_C    (WS_U+8)          // [8]     c[s] = keys[s] . br
#define WS_ZCNT (D*NS + 64 + 8 + 1)   // zero range starting at WS_W0

// ---------------------------------------------------------------- K0: zero accumulators
__global__ void k0_zero(float* ws) {
    int i = blockIdx.x * 256 + threadIdx.x;
    if (i < WS_ZCNT) ws[WS_W0 + i] = 0.0f;
}

// ---------------------------------------------------------------- K1: precompute M, S, u, c
__global__ __launch_bounds__(256)
void k1_pre(const float* __restrict__ keys, const float* __restrict__ Wr,
            const float* __restrict__ br,   const float* __restrict__ Wc,
            const float* __restrict__ state,const float* __restrict__ fast,
            float* __restrict__ ws) {
    const int bid = blockIdx.x, tid = threadIdx.x;
    if (bid < 8) {
        // M[k][s]: 2048 k values, 8 blocks x 256 threads
        __shared__ float keysl[NS * D];
        for (int i = tid; i < NS * D; i += 256) keysl[i] = keys[i];
        __syncthreads();
        const int k = bid * 256 + tid;
        float acc[NS];
#pragma unroll
        for (int s = 0; s < NS; ++s) acc[s] = 0.0f;
        for (int d = 0; d < D; ++d) {
            const float w = Wr[(size_t)d * D + k];
#pragma unroll
            for (int s = 0; s < NS; ++s) acc[s] = fmaf(keysl[s * D + d], w, acc[s]);
        }
#pragma unroll
        for (int s = 0; s < NS; ++s) ws[WS_M + k * NS + s] = acc[s];
    } else if (bid < 16) {
        // S = state + fast
        const int base = (bid - 8) * 2048;
#pragma unroll
        for (int j = 0; j < 8; ++j) {
            const int i = base + tid + j * 256;
            ws[WS_S + i] = state[i] + fast[i];
        }
    } else {
        // u[s], c[s]: one wave per slot
        const int s = tid >> 5, lane = tid & 31;
        float su = 0.0f, sc = 0.0f;
        for (int k = lane; k < D; k += 32) {
            su += (state[s * D + k] + fast[s * D + k]) * Wc[D + k]; // Wc[0][D+k]
            sc += keys[s * D + k] * br[k];
        }
        for (int off = 16; off > 0; off >>= 1) {
            su += __shfl_down(su, off, 32);
            sc += __shfl_down(sc, off, 32);
        }
        if (lane == 0) { ws[WS_U + s] = su; ws[WS_C + s] = sc; }
    }
}

// ---------------------------------------------------------------- K2: fused main pass (32 rows / block)
__global__ __launch_bounds__(256)
void k2_main(const float* __restrict__ h, const float* __restrict__ nm,
             const float* __restrict__ Wc, const float* __restrict__ bc,
             float* __restrict__ ws, float* __restrict__ out) {
    __shared__ float attnL[32 * NS];
    __shared__ float gateP;
    const int tid = threadIdx.x;
    const int rowbase = blockIdx.x * 32;
    if (tid == 0) gateP = 0.0f;

    // ---- phase 1: logits = h @ M + c, softmax over 8 slots (thread = (row, slot))
    const int row = tid >> 3, s = tid & 7;
    const float* hrow = h + (size_t)(rowbase + row) * D;
    float acc = 0.0f, gacc = 0.0f;
    for (int k = 0; k < D; k += 4) {
        const float4 hv = *(const float4*)(hrow + k);
        acc = fmaf(hv.x, ws[WS_M + (k + 0) * NS + s], acc);
        acc = fmaf(hv.y, ws[WS_M + (k + 1) * NS + s], acc);
        acc = fmaf(hv.z, ws[WS_M + (k + 2) * NS + s], acc);
        acc = fmaf(hv.w, ws[WS_M + (k + 3) * NS + s], acc);
        if (s == 0)
            gacc += hv.x * Wc[k] + hv.y * Wc[k + 1] + hv.z * Wc[k + 2] + hv.w * Wc[k + 3];
    }
    const float logit = acc + ws[WS_C + s];
    float mx = logit;
    for (int off = 1; off < 8; off <<= 1) mx = fmaxf(mx, __shfl_xor(mx, off, 8));
    const float e = expf(logit - mx);
    float sum = e;
    for (int off = 1; off < 8; off <<= 1) sum += __shfl_xor(sum, off, 8);
    const float attn = e / sum;
    attnL[row * NS + s] = attn;
    out[(size_t)B * D + (size_t)(rowbase + row) * NS + s] = attn; // attn output
    __syncthreads();

    // ---- phase 2: gate sigmoid sum, Gram, colsum
    if (s == 0) {
        float g = gacc + bc[0];
#pragma unroll
        for (int ss = 0; ss < NS; ++ss) g = fmaf(attnL[row * NS + ss], ws[WS_U + ss], g);
        atomicAdd(&gateP, 1.0f / (1.0f + expf(-g)));
    }
    if (tid < 64) {
        const int sA = tid >> 3, sB = tid & 7;
        float gs = 0.0f;
        for (int r = 0; r < 32; ++r) gs = fmaf(attnL[r * NS + sA], attnL[r * NS + sB], gs);
        atomicAdd(&ws[WS_GRAM + sA * NS + sB], gs);
    } else if (tid < 72) {
        const int ss = tid - 64;
        float cs = 0.0f;
        for (int r = 0; r < 32; ++r) cs += attnL[r * NS + ss];
        atomicAdd(&ws[WS_COL + ss], cs);
    }
    __syncthreads();
    if (tid == 0) atomicAdd(&ws[WS_GATE], gateP);

    // ---- phase 3a: read = attn @ S via V_WMMA_F32_16X16X4_F32 (exact fp32)
    // wave handles one 16-row tile; K=8 split into two K=4 WMMA steps.
    const int wvid = tid >> 5, lane = tid & 31;
    const int rt = wvid & 1;          // which 16-row half of the 32-row block
    const int mrow = lane & 15, hi = lane >> 4, n = lane & 15;
    v2f a0, a1;   // A layout (32-bit 16x4): lane=row; vgpr0=K0|K2, vgpr1=K1|K3
    a0.x = attnL[(rt * 16 + mrow) * NS + hi * 2 + 0];
    a0.y = attnL[(rt * 16 + mrow) * NS + hi * 2 + 1];
    a1.x = attnL[(rt * 16 + mrow) * NS + 4 + hi * 2 + 0];
    a1.y = attnL[(rt * 16 + mrow) * NS + 4 + hi * 2 + 1];
    const float* Sg = ws + WS_S;
    for (int dt = (wvid >> 1); dt < D / 16; dt += 4) {
        const int d0 = dt * 16;
        v2f b0, b1;   // B layout (4x16): lane=col; vgpr0=K0|K2, vgpr1=K1|K3
        b0.x = Sg[(hi * 2 + 0) * D + d0 + n];
        b0.y = Sg[(hi * 2 + 1) * D + d0 + n];
        b1.x = Sg[(4 + hi * 2 + 0) * D + d0 + n];
        b1.y = Sg[(4 + hi * 2 + 1) * D + d0 + n];
        v8f cc = {};
        cc = __builtin_amdgcn_wmma_f32_16x16x4_f32(false, a0, false, b0, (short)0, cc, false, false);
        cc = __builtin_amdgcn_wmma_f32_16x16x4_f32(false, a1, false, b1, (short)0, cc, false, false);
#pragma unroll
        for (int j = 0; j < 8; ++j) {   // C/D layout: vgpr j -> row j (lanes<16) / j+8 (lanes>=16)
            const int rm = j + hi * 8;
            out[(size_t)(rowbase + rt * 16 + rm) * D + d0 + n] = cc[j];
        }
    }

    // ---- phase 3b: W0 += attn.T @ (h + neighbor_msg), register-accumulated
    float accw[8][NS];
#pragma unroll
    for (int j = 0; j < 8; ++j)
#pragma unroll
        for (int ss = 0; ss < NS; ++ss) accw[j][ss] = 0.0f;
    for (int r = 0; r < 32; ++r) {
        float a8[NS];
#pragma unroll
        for (int ss = 0; ss < NS; ++ss) a8[ss] = attnL[r * NS + ss];
        const float* hr = h  + (size_t)(rowbase + r) * D;
        const float* nr = nm + (size_t)(rowbase + r) * D;
#pragma unroll
        for (int j = 0; j < 8; ++j) {
            const float hv = hr[tid + j * 256] + nr[tid + j * 256];
#pragma unroll
            for (int ss = 0; ss < NS; ++ss) accw[j][ss] = fmaf(a8[ss], hv, accw[j][ss]);
        }
    }
#pragma unroll
    for (int j = 0; j < 8; ++j) {
        const int k = tid + j * 256;
#pragma unroll
        for (int ss = 0; ss < NS; ++ss) atomicAdd(&ws[WS_W0 + ss * D + k], accw[j][ss]);
    }
}

// ---------------------------------------------------------------- K3: delta GEMV + state/fast update + alpha
__global__ __launch_bounds__(128)
void k3_final(const float* __restrict__ Ww, const float* __restrict__ bw,
              const float* __restrict__ state, const float* __restrict__ fast,
              const float* __restrict__ ws, float* __restrict__ out) {
    __shared__ float AWl[NS * D2];   // [8][4096] = 128 KB: [W0 | Gram@S]
    const int tid = threadIdx.x;
    for (int i = tid; i < NS * D; i += 128) {
        const int s = i / D, k = i % D;
        AWl[s * D2 + k] = ws[WS_W0 + i];
    }
    for (int i = tid; i < NS * D; i += 128) {
        const int s = i / D, k = i % D;
        float v = 0.0f;
#pragma unroll
        for (int sp = 0; sp < NS; ++sp)
            v = fmaf(ws[WS_GRAM + s * NS + sp], ws[WS_S + sp * D + k], v);
        AWl[s * D2 + D + k] = v;
    }
    __syncthreads();

    const float mean = ws[WS_GATE] * (1.0f / (float)B);
    float alpha = 0.02f + 0.98f * mean;          // ALPHA_MIN + (1-ALPHA_MIN)*mean
    alpha = fminf(fmaxf(alpha, 0.0f), 1.0f);     // clip; ALPHA_BOOST == 1

    const size_t o_alpha = (size_t)B * D + (size_t)B * NS;
    if (blockIdx.x == 0 && tid == 0) out[o_alpha] = alpha;

    const int d = blockIdx.x * 128 + tid;
    float acc[NS];
#pragma unroll
    for (int s = 0; s < NS; ++s) acc[s] = 0.0f;
    const float* wr = Ww + (size_t)d * D2;
    for (int k = 0; k < D2; k += 4) {
        const float4 wv = *(const float4*)(wr + k);
#pragma unroll
        for (int s = 0; s < NS; ++s) {
            const float4 av = *(const float4*)(&AWl[s * D2 + k]);
            acc[s] += av.x * wv.x + av.y * wv.y + av.z * wv.z + av.w * wv.w;
        }
    }
    const float bwv = bw[d];
    const size_t o_state = o_alpha + 1;
    const size_t o_fast  = o_state + (size_t)NS * D;
#pragma unroll
    for (int s = 0; s < NS; ++s) {
        const float delta = acc[s] + ws[WS_COL + s] * bwv;
        out[o_state + s * D + d] = (1.0f - alpha) * state[s * D + d] + alpha * delta;
        out[o_fast  + s * D + d] = 0.95f * fast[s * D + d] + 0.05f * delta;
    }
}

// ----------------------------------------------------------------
extern "C" void kernel_launch(void* const* d_in, const int* in_sizes, int n_in,
                              void* d_out, int out_size, void* d_ws, size_t ws_size,
                              hipStream_t stream) {
    const float* h     = (const float*)d_in[0];
    const float* nmsg  = (const float*)d_in[1];
    const float* keys  = (const float*)d_in[2];
    const float* Wr    = (const float*)d_in[3];
    const float* br    = (const float*)d_in[4];
    const float* Wc    = (const float*)d_in[5];
    const float* bc    = (const float*)d_in[6];
    const float* Ww    = (const float*)d_in[7];
    const float* bw    = (const float*)d_in[8];
    const float* state = (const float*)d_in[9];
    const float* fast  = (const float*)d_in[10];
    float* out = (float*)d_out;
    float* ws  = (float*)d_ws;

    k0_zero <<<(WS_ZCNT + 255) / 256, 256, 0, stream>>>(ws);
    k1_pre  <<<17,  256, 0, stream>>>(keys, Wr, br, Wc, state, fast, ws);
    k2_main <<<B / 32, 256, 0, stream>>>(h, nmsg, Wc, bc, ws, out);
    k3_final<<<D / 128, 128, 0, stream>>>(Ww, bw, state, fast, ws, out);
}